// AdaptivelyScaledCALayer_41394894799126
// MI455X (gfx1250) — compile-verified
//
#include <hip/hip_runtime.h>
#include <hip/hip_bf16.h>
#include <math.h>

// Problem constants (match reference)
#define BB_ 16
#define CC_ 256
#define HW_ 16384           // 128*128
#define NCH (BB_ * CC_)     // 4096 channels total

typedef __attribute__((ext_vector_type(2))) float v2f;
typedef __attribute__((ext_vector_type(8))) float v8f;

// ---------------------------------------------------------------------------
// Kernel 1: per-(b,c) spatial sum / sumsq reduction -> mean, std
// One 256-thread block per channel; each thread streams 16 float4 (b128) loads.
// ---------------------------------------------------------------------------
__global__ void __launch_bounds__(256) reduce_kernel(const float* __restrict__ x,
                                                     float* __restrict__ g_mean,
                                                     float* __restrict__ g_std) {
    const int bc  = blockIdx.x;       // 0..4095
    const int tid = threadIdx.x;      // 0..255
    const float4* __restrict__ x4 = (const float4*)x + (size_t)bc * (HW_ / 4);

    float s = 0.f, sq = 0.f;
#pragma unroll
    for (int k = 0; k < 16; ++k) {
        float4 v = x4[k * 256 + tid];
        s  += (v.x + v.y) + (v.z + v.w);
        sq += (v.x * v.x + v.y * v.y) + (v.z * v.z + v.w * v.w);
    }

    __shared__ float ls[256];
    __shared__ float lq[256];
    ls[tid] = s;
    lq[tid] = sq;
    __syncthreads();
    for (int off = 128; off > 0; off >>= 1) {
        if (tid < off) {
            ls[tid] += ls[tid + off];
            lq[tid] += lq[tid + off];
        }
        __syncthreads();
    }
    if (tid == 0) {
        const float inv = 1.0f / (float)HW_;
        float m   = ls[0] * inv;
        float var = lq[0] * inv - m * m;
        g_mean[bc] = m;
        g_std[bc]  = sqrtf(fmaxf(var, 0.0f));
    }
}

// ---------------------------------------------------------------------------
// WMMA helpers: D = A(16xK, LDS) * W^T(KxN, W is [Ntot x K] in global) + C
// f32 path: V_WMMA_F32_16X16X4_F32, K-step = 4.
// A 16x4 frag layout: lanes 0-15 -> K={0,1}, lanes 16-31 -> K={2,3}, M=lane&15.
// B 4x16 frag layout: symmetric, N=lane&15.
// C/D: VGPR r holds M = r + 8*(lane>=16), N = lane&15.
// ---------------------------------------------------------------------------
__device__ __forceinline__ v8f gemm_tile(const float* __restrict__ As, int lda, int acol0,
                                         const float* __restrict__ W,  int ldw, int ncol0,
                                         int K, int lane) {
    const int row = lane & 15;
    const int kh  = (lane >> 4) * 2;
    const int col = ncol0 + row;
    v8f acc = {};
    for (int k0 = 0; k0 < K; k0 += 4) {
        v2f a, b;
        a.x = As[row * lda + acol0 + k0 + kh];
        a.y = As[row * lda + acol0 + k0 + kh + 1];
        b.x = W[col * ldw + k0 + kh];        // B[k][n] = W[n][k]
        b.y = W[col * ldw + k0 + kh + 1];
        acc = __builtin_amdgcn_wmma_f32_16x16x4_f32(
            /*neg_a=*/false, a, /*neg_b=*/false, b,
            /*c_mod=*/(short)0, acc, /*reuse_a=*/false, /*reuse_b=*/false);
    }
    return acc;
}

// MODE: 0 = bias only, 1 = bias+relu, 2 = bias+sigmoid
template <int MODE>
__device__ __forceinline__ void emit_tile(float* __restrict__ Ds, int ldd, int dcol0,
                                          v8f acc, float bval, int lane) {
    const int row  = lane & 15;
    const int half = lane >> 4;
#pragma unroll
    for (int r = 0; r < 8; ++r) {
        float v = acc[r] + bval;
        if (MODE == 1) v = fmaxf(v, 0.0f);
        if (MODE == 2) v = 1.0f / (1.0f + __expf(-v));
        Ds[(r + 8 * half) * ldd + dcol0 + row] = v;
    }
}

// ---------------------------------------------------------------------------
// Kernel 2: the whole SE MLP chain in one 256-thread (8-wave) workgroup.
// All GEMMs via V_WMMA_F32_16X16X4_F32; intermediates in LDS.
// ---------------------------------------------------------------------------
__global__ void __launch_bounds__(256) se_mlp_kernel(
    const float* __restrict__ g_std, const float* __restrict__ g_mean,
    const float* __restrict__ w1s, const float* __restrict__ b1s,
    const float* __restrict__ w2s, const float* __restrict__ b2s,
    const float* __restrict__ w1m, const float* __restrict__ b1m,
    const float* __restrict__ w2m, const float* __restrict__ b2m,
    const float* __restrict__ wb,  const float* __restrict__ bb,
    const float* __restrict__ w1f, const float* __restrict__ b1f,
    const float* __restrict__ w2f, const float* __restrict__ b2f,
    float* __restrict__ g_mask) {

    __shared__ float sCat[BB_][2 * CC_];   // concat buffer: [16][512] (32 KB)
    __shared__ float sFus[BB_][CC_];       // bottleneck out: [16][256] (16 KB)
    __shared__ float sHs[BB_][16];         // squeeze-std hidden
    __shared__ float sHm[BB_][16];         // squeeze-mean hidden
    __shared__ float sHf[BB_][16];         // squeeze-final hidden

    const int tid  = threadIdx.x;
    const int lane = tid & 31;
    const int wave = tid >> 5;             // 0..7
    const int row  = lane & 15;

    // Load descriptors: std into cols [0,256), mean into cols [256,512)
    for (int i = tid; i < BB_ * CC_; i += 256) {
        int b = i >> 8, c = i & 255;
        sCat[b][c]       = g_std[i];
        sCat[b][CC_ + c] = g_mean[i];
    }
    __syncthreads();

    // Stage B: squeeze hidden layers  h = relu(desc @ W1^T + b1)  [16x16]
    if (wave == 0) {
        v8f acc = gemm_tile(&sCat[0][0], 2 * CC_, 0,   w1s, CC_, 0, CC_, lane);
        emit_tile<1>(&sHs[0][0], 16, 0, acc, b1s[row], lane);
    } else if (wave == 1) {
        v8f acc = gemm_tile(&sCat[0][0], 2 * CC_, CC_, w1m, CC_, 0, CC_, lane);
        emit_tile<1>(&sHm[0][0], 16, 0, acc, b1m[row], lane);
    }
    __syncthreads();

    // Stage C: expand  se = h @ W2^T + b2  [16x256]; overwrite concat buffer
    for (int t = wave * 2; t < wave * 2 + 2; ++t) {
        const int n0 = t * 16;
        v8f as = gemm_tile(&sHs[0][0], 16, 0, w2s, 16, n0, 16, lane);
        v8f am = gemm_tile(&sHm[0][0], 16, 0, w2m, 16, n0, 16, lane);
        emit_tile<0>(&sCat[0][0], 2 * CC_, n0,       as, b2s[n0 + row], lane);
        emit_tile<0>(&sCat[0][0], 2 * CC_, CC_ + n0, am, b2m[n0 + row], lane);
    }
    __syncthreads();

    // Stage D: bottleneck  fused = relu(cat @ Wb^T + bb)  [16x256], K=512
    for (int t = wave * 2; t < wave * 2 + 2; ++t) {
        const int n0 = t * 16;
        v8f acc = gemm_tile(&sCat[0][0], 2 * CC_, 0, wb, 2 * CC_, n0, 2 * CC_, lane);
        emit_tile<1>(&sFus[0][0], CC_, n0, acc, bb[n0 + row], lane);
    }
    __syncthreads();

    // Stage E: final squeeze hidden  hf = relu(fused @ W1f^T + b1f)  [16x16]
    if (wave == 0) {
        v8f acc = gemm_tile(&sFus[0][0], CC_, 0, w1f, CC_, 0, CC_, lane);
        emit_tile<1>(&sHf[0][0], 16, 0, acc, b1f[row], lane);
    }
    __syncthreads();

    // Stage F: final expand + sigmoid -> mask [16x256] to global ws
    for (int t = wave * 2; t < wave * 2 + 2; ++t) {
        const int n0 = t * 16;
        v8f acc = gemm_tile(&sHf[0][0], 16, 0, w2f, 16, n0, 16, lane);
        emit_tile<2>(g_mask, CC_, n0, acc, b2f[n0 + row], lane);
    }
}

// ---------------------------------------------------------------------------
// Kernel 3: out = x * mask[b,c]   (pure streaming, b128 load/store)
// Each block covers 1024 floats -> single channel per block, mask is a
// wave-uniform scalar load.
// ---------------------------------------------------------------------------
__global__ void __launch_bounds__(256) scale_kernel(const float* __restrict__ x,
                                                    const float* __restrict__ mask,
                                                    float* __restrict__ out) {
    const size_t i4 = (size_t)blockIdx.x * 256 + threadIdx.x;  // float4 index
    const int bc = (int)(i4 >> 12);                            // 4096 float4 / channel
    const float m = mask[bc];
    float4 v = ((const float4*)x)[i4];
    v.x *= m; v.y *= m; v.z *= m; v.w *= m;
    ((float4*)out)[i4] = v;
}

// ---------------------------------------------------------------------------
extern "C" void kernel_launch(void* const* d_in, const int* in_sizes, int n_in,
                              void* d_out, int out_size, void* d_ws, size_t ws_size,
                              hipStream_t stream) {
    const float* x   = (const float*)d_in[0];
    const float* w1s = (const float*)d_in[1];
    const float* b1s = (const float*)d_in[2];
    const float* w2s = (const float*)d_in[3];
    const float* b2s = (const float*)d_in[4];
    const float* w1m = (const float*)d_in[5];
    const float* b1m = (const float*)d_in[6];
    const float* w2m = (const float*)d_in[7];
    const float* b2m = (const float*)d_in[8];
    const float* wb  = (const float*)d_in[9];
    const float* bb  = (const float*)d_in[10];
    const float* w1f = (const float*)d_in[11];
    const float* b1f = (const float*)d_in[12];
    const float* w2f = (const float*)d_in[13];
    const float* b2f = (const float*)d_in[14];

    float* out    = (float*)d_out;
    float* ws     = (float*)d_ws;
    float* g_mean = ws;                 // [4096]
    float* g_std  = ws + NCH;           // [4096]
    float* g_mask = ws + 2 * NCH;       // [4096]

    // Pass 1: 256 MB read -> descriptors
    reduce_kernel<<<NCH, 256, 0, stream>>>(x, g_mean, g_std);

    // Tiny WMMA MLP: one workgroup
    se_mlp_kernel<<<1, 256, 0, stream>>>(g_std, g_mean,
                                         w1s, b1s, w2s, b2s,
                                         w1m, b1m, w2m, b2m,
                                         wb, bb, w1f, b1f, w2f, b2f,
                                         g_mask);

    // Pass 2: 256 MB read + 256 MB write
    const int totalF4  = (BB_ * CC_ * HW_) / 4;   // 16,777,216
    const int nblocks  = totalF4 / 256;           // 65,536
    scale_kernel<<<nblocks, 256, 0, stream>>>(x, g_mask, out);
}